// GATModel_13804024889626
// MI455X (gfx1250) — compile-verified
//
#include <hip/hip_runtime.h>
#include <math.h>

#define NN 50000       // nodes
#define NE 800000      // edges
#define IN_F 64        // input feats
#define H 8            // heads
#define O 32           // hidden per head
#define HO 256         // H*O
#define NEG_SLOPE 0.2f

typedef __attribute__((ext_vector_type(2))) float v2f;
typedef __attribute__((ext_vector_type(8))) float v8f;

// ---- order-preserving float<->uint mapping for atomicMax-based segment max ----
__device__ __forceinline__ unsigned flipf(float f) {
  unsigned u = __float_as_uint(f);
  return (u & 0x80000000u) ? ~u : (u | 0x80000000u);
}
__device__ __forceinline__ float unflipf(unsigned u) {
  unsigned b = (u & 0x80000000u) ? (u ^ 0x80000000u) : ~u;
  return __uint_as_float(b);
}
__device__ __forceinline__ float lrelu(float x) { return x > 0.f ? x : NEG_SLOPE * x; }

// ---------------- zero scratch region ----------------
__global__ void k_zero(unsigned* __restrict__ p, long long n) {
  long long i = (long long)blockIdx.x * blockDim.x + threadIdx.x;
  long long stride = (long long)gridDim.x * blockDim.x;
  for (; i < n; i += stride) p[i] = 0u;
}

// ---------------- Layer 1 GEMM: feat1[NN,HO] = X[NN,IN_F] @ W1[IN_F,HO] via fp32 WMMA ----------------
// One wave per 16x16 output tile. 3125 row tiles x 16 col tiles = 50000 waves.
__global__ void k_gemm1(const float* __restrict__ X, const float* __restrict__ W,
                        float* __restrict__ F) {
  const int lane = threadIdx.x & 31;
  const int tile = (blockIdx.x * blockDim.x + threadIdx.x) >> 5; // global wave id
  const int rowTile = tile >> 4;   // / 16 col-tiles
  const int colTile = tile & 15;
  if (rowTile >= NN / 16) return;  // uniform per wave
  const int half = lane >> 4;      // 0: lanes 0-15, 1: lanes 16-31
  const int l = lane & 15;
  const int row0 = rowTile * 16;
  const int col0 = colTile * 16;

  v8f c = {};
  // A (16x4 f32): lane half selects K pair {0,1} vs {2,3}; row = row0 + l
  const float* xrow = X + (size_t)(row0 + l) * IN_F + 2 * half;
  // B (4x16 f32): mirrored layout; col = col0 + l
  const float* wcol = W + (size_t)(2 * half) * HO + col0 + l;

#pragma unroll
  for (int k = 0; k < IN_F; k += 4) {
    v2f a = *(const v2f*)(xrow + k);          // K=k+2*half, k+2*half+1
    v2f b;
    b.x = wcol[(size_t)k * HO];               // K=k+2*half
    b.y = wcol[(size_t)(k + 1) * HO];         // K=k+2*half+1
    c = __builtin_amdgcn_wmma_f32_16x16x4_f32(
        /*neg_a=*/false, a, /*neg_b=*/false, b,
        /*c_mod=*/(short)0, c, /*reuse_a=*/false, /*reuse_b=*/false);
  }
  // C/D: VGPR r -> M = r + 8*half, N = l
  float* fp = F + (size_t)(row0 + 8 * half) * HO + col0 + l;
#pragma unroll
  for (int r = 0; r < 8; ++r) fp[(size_t)r * HO] = c[r];
}

// ---------------- attention coefficients el/er per (node, head) ----------------
__global__ void k_attn1(const float* __restrict__ feat1, const float* __restrict__ al,
                        const float* __restrict__ ar, float* __restrict__ el,
                        float* __restrict__ er) {
  int t = blockIdx.x * blockDim.x + threadIdx.x;
  if (t >= NN * H) return;
  int n = t >> 3, h = t & 7;
  const float* f = feat1 + (size_t)n * HO + h * O;
  const float* pl = al + h * O;
  const float* pr = ar + h * O;
  float sl = 0.f, sr = 0.f;
#pragma unroll
  for (int o = 0; o < O; ++o) { sl += f[o] * pl[o]; sr += f[o] * pr[o]; }
  el[t] = sl;
  er[t] = sr;
}

// ---------------- layer-1 edge passes ----------------
__global__ void k_edge1_a(const int* __restrict__ src, const int* __restrict__ dst,
                          const float* __restrict__ el, const float* __restrict__ er,
                          float* __restrict__ ews, unsigned* __restrict__ m1) {
  int t = blockIdx.x * blockDim.x + threadIdx.x;
  if (t >= NE * H) return;
  int e = t >> 3, h = t & 7;
  int s = src[e], d = dst[e];
  float v = lrelu(el[s * H + h] + er[d * H + h]);
  ews[t] = v;
  atomicMax(&m1[d * H + h], flipf(v));
}

__global__ void k_edge1_b(const int* __restrict__ dst, const unsigned* __restrict__ m1,
                          float* __restrict__ ews, float* __restrict__ s1) {
  int t = blockIdx.x * blockDim.x + threadIdx.x;
  if (t >= NE * H) return;
  int e = t >> 3, h = t & 7;
  int d = dst[e];
  float ex = __expf(ews[t] - unflipf(m1[d * H + h]));
  ews[t] = ex;
  atomicAdd(&s1[d * H + h], ex);
}

// one thread per (edge, h, o): alpha-weighted scatter of feat1[src] into out1[dst]
__global__ void k_edge1_c(const int* __restrict__ src, const int* __restrict__ dst,
                          const float* __restrict__ ews, const float* __restrict__ s1,
                          const float* __restrict__ feat1, float* __restrict__ out1) {
  int t = blockIdx.x * blockDim.x + threadIdx.x; // < NE*HO = 204.8M (fits int)
  int e = t >> 8;
  int c = t & 255;
  int h = c >> 5;
  int s = src[e], d = dst[e];
  float alpha = ews[e * H + h] / s1[d * H + h];
  atomicAdd(&out1[(size_t)d * HO + c], feat1[(size_t)s * HO + c] * alpha);
}

// out1 = relu(out1 + b1)
__global__ void k_bias_relu1(float* __restrict__ out1, const float* __restrict__ b1) {
  int t = blockIdx.x * blockDim.x + threadIdx.x;
  if (t >= NN * HO) return;
  float v = out1[t] + b1[t & 255];
  out1[t] = v > 0.f ? v : 0.f;
}

// ---------------- layer 2: scalar projection + attention coefs ----------------
__global__ void k_node2(const float* __restrict__ h1, const float* __restrict__ W2,
                        const float* __restrict__ al2, const float* __restrict__ ar2,
                        float* __restrict__ h2, float* __restrict__ el2,
                        float* __restrict__ er2) {
  int n = blockIdx.x * blockDim.x + threadIdx.x;
  if (n >= NN) return;
  const float* p = h1 + (size_t)n * HO;
  float acc = 0.f;
#pragma unroll 8
  for (int i = 0; i < HO; ++i) acc += p[i] * W2[i];
  h2[n] = acc;
  el2[n] = acc * al2[0];
  er2[n] = acc * ar2[0];
}

__global__ void k_edge2_a(const int* __restrict__ src, const int* __restrict__ dst,
                          const float* __restrict__ el, const float* __restrict__ er,
                          float* __restrict__ ews, unsigned* __restrict__ m2) {
  int e = blockIdx.x * blockDim.x + threadIdx.x;
  if (e >= NE) return;
  float v = lrelu(el[src[e]] + er[dst[e]]);
  ews[e] = v;
  atomicMax(&m2[dst[e]], flipf(v));
}

__global__ void k_edge2_b(const int* __restrict__ dst, const unsigned* __restrict__ m2,
                          float* __restrict__ ews, float* __restrict__ s2) {
  int e = blockIdx.x * blockDim.x + threadIdx.x;
  if (e >= NE) return;
  int d = dst[e];
  float ex = __expf(ews[e] - unflipf(m2[d]));
  ews[e] = ex;
  atomicAdd(&s2[d], ex);
}

__global__ void k_edge2_c(const int* __restrict__ src, const int* __restrict__ dst,
                          const float* __restrict__ ews, const float* __restrict__ s2,
                          const float* __restrict__ h2, float* __restrict__ out2) {
  int e = blockIdx.x * blockDim.x + threadIdx.x;
  if (e >= NE) return;
  int d = dst[e];
  float alpha = ews[e] / s2[d];
  atomicAdd(&out2[d], h2[src[e]] * alpha);
}

__global__ void k_final(const float* __restrict__ out2, const float* __restrict__ b2,
                        float* __restrict__ out) {
  int n = blockIdx.x * blockDim.x + threadIdx.x;
  if (n >= NN) return;
  float v = out2[n] + b2[0];
  out[n] = 1.f / (1.f + __expf(-v));
}

// ======================================================================
extern "C" void kernel_launch(void* const* d_in, const int* in_sizes, int n_in,
                              void* d_out, int out_size, void* d_ws, size_t ws_size,
                              hipStream_t stream) {
  const float* X   = (const float*)d_in[0];   // [NN, IN_F]
  const int*   src = (const int*)d_in[1];     // [NE]
  const int*   dst = (const int*)d_in[2];     // [NE]
  // d_in[3] = edge_types (unused by reference)
  const float* W1  = (const float*)d_in[4];   // [IN_F, H, O] = [64,256]
  const float* al1 = (const float*)d_in[5];   // [H,O]
  const float* ar1 = (const float*)d_in[6];
  const float* b1  = (const float*)d_in[7];
  const float* W2  = (const float*)d_in[8];   // [HO,1,1]
  const float* al2 = (const float*)d_in[9];
  const float* ar2 = (const float*)d_in[10];
  const float* b2  = (const float*)d_in[11];
  float* out = (float*)d_out;                 // [NN]

  // ---- workspace layout (floats/uints, 4B each) ----
  float* ws = (float*)d_ws;
  size_t off = 0;
  float*    feat1 = ws + off; off += (size_t)NN * HO;       // 12.8M
  size_t zero_begin = off;                                  // zero region start
  float*    out1  = ws + off; off += (size_t)NN * HO;       // 12.8M
  unsigned* m1    = (unsigned*)(ws + off); off += (size_t)NN * H;
  float*    s1    = ws + off; off += (size_t)NN * H;
  float*    out2  = ws + off; off += NN;
  unsigned* m2    = (unsigned*)(ws + off); off += NN;
  float*    s2    = ws + off; off += NN;
  size_t zero_count = off - zero_begin;                     // contiguous zero region
  float*    el1   = ws + off; off += (size_t)NN * H;
  float*    er1   = ws + off; off += (size_t)NN * H;
  float*    ews   = ws + off; off += (size_t)NE * H;        // reused (first NE) in layer 2
  float*    h2    = ws + off; off += NN;
  float*    el2   = ws + off; off += NN;
  float*    er2   = ws + off; off += NN;
  (void)ws_size; (void)n_in; (void)in_sizes; (void)out_size;

  const int B = 256;

  // 0) zero accumulators (out1, m1, s1, out2, m2, s2) — flipped-max sentinel 0 == -inf
  k_zero<<<16384, B, 0, stream>>>((unsigned*)(ws + zero_begin), (long long)zero_count);

  // 1) layer-1 GEMM via WMMA: 50000 waves, 8 waves/block
  k_gemm1<<<(NN / 16) * 16 / 8, B, 0, stream>>>(X, W1, feat1);

  // 2) el/er per (node, head)
  k_attn1<<<(NN * H + B - 1) / B, B, 0, stream>>>(feat1, al1, ar1, el1, er1);

  // 3) edge softmax + aggregation, layer 1
  k_edge1_a<<<(NE * H + B - 1) / B, B, 0, stream>>>(src, dst, el1, er1, ews, m1);
  k_edge1_b<<<(NE * H + B - 1) / B, B, 0, stream>>>(dst, m1, ews, s1);
  k_edge1_c<<<NE * HO / B, B, 0, stream>>>(src, dst, ews, s1, feat1, out1);

  // 4) bias + relu
  k_bias_relu1<<<(NN * HO + B - 1) / B, B, 0, stream>>>(out1, b1);

  // 5) layer-2 scalar projection
  k_node2<<<(NN + B - 1) / B, B, 0, stream>>>(out1, W2, al2, ar2, h2, el2, er2);

  // 6) edge softmax + aggregation, layer 2 (H=1, O=1)
  k_edge2_a<<<(NE + B - 1) / B, B, 0, stream>>>(src, dst, el2, er2, ews, m2);
  k_edge2_b<<<(NE + B - 1) / B, B, 0, stream>>>(dst, m2, ews, s2);
  k_edge2_c<<<(NE + B - 1) / B, B, 0, stream>>>(src, dst, ews, s2, h2, out2);

  // 7) sigmoid
  k_final<<<(NN + B - 1) / B, B, 0, stream>>>(out2, b2, out);
}